// MeanAggregator_66245575573681
// MI455X (gfx1250) — compile-verified
//
#include <hip/hip_runtime.h>

typedef float v2f __attribute__((ext_vector_type(2)));
typedef float v4f __attribute__((ext_vector_type(4)));
typedef float v8f __attribute__((ext_vector_type(8)));

#define N_NODES 100000
#define E_EDGES 3200000
#define DIM     128

// ---------------------------------------------------------------------------
// Kernel 1: h = x @ W + b using V_WMMA_F32_16X16X4_F32 (fp32 tensor path).
// One wave per 16x16 output tile; 8 waves per block cover 16 rows x 128 cols.
// A (16x4 f32): lanes 0-15 hold K=0,1 (2 VGPRs), lanes 16-31 hold K=2,3.
// B (4x16 f32): lanes 0-15 = n for K=0,1; lanes 16-31 = n for K=2,3.
// C/D (16x16 f32, 8 VGPRs): VGPR r, lane L -> row M = r + 8*(L>>4), col N = L&15.
// h is stored with default (RT) temporal hints so it stays resident in the
// 192MB L2 for the gather stage (h = 51.2MB).
// ---------------------------------------------------------------------------
__global__ __launch_bounds__(256) void gemm_bias_wmma(
    const float* __restrict__ x, const float* __restrict__ W,
    const float* __restrict__ bias, float* __restrict__ h)
{
    const int lane  = threadIdx.x & 31;
    const int wave  = threadIdx.x >> 5;          // 0..7 -> N tile
    const int m0    = blockIdx.x * 16;
    const int n0    = wave * 16;

    const int l15   = lane & 15;
    const int khalf = (lane >> 4) * 2;           // 0 or 2

    v8f acc = {};
    const float* xrow = x + (size_t)(m0 + l15) * DIM;

#pragma unroll 4
    for (int kk = 0; kk < DIM; kk += 4) {
        const int kb = kk + khalf;
        v2f a;
        a.x = xrow[kb];
        a.y = xrow[kb + 1];
        v2f bf;
        bf.x = W[(size_t)kb       * DIM + n0 + l15];
        bf.y = W[(size_t)(kb + 1) * DIM + n0 + l15];
        acc = __builtin_amdgcn_wmma_f32_16x16x4_f32(
            /*neg_a=*/false, a, /*neg_b=*/false, bf,
            /*c_mod=*/(short)0, acc, /*reuse_a=*/false, /*reuse_b=*/false);
    }

    const float bv    = bias[n0 + l15];
    const int   mbase = m0 + 8 * (lane >> 4);
    float* hcol = h + (size_t)mbase * DIM + n0 + l15;
#pragma unroll
    for (int r = 0; r < 8; ++r)
        hcol[(size_t)r * DIM] = acc[r] + bv;
}

// ---------------------------------------------------------------------------
// Kernel 2: row_ptr[i] = lower_bound(row, i).  row is sorted int64, so a
// 22-step binary search per node gives a CSR without any atomics.
// ---------------------------------------------------------------------------
__global__ __launch_bounds__(256) void rowptr_kernel(
    const long long* __restrict__ row, int* __restrict__ rptr)
{
    const int i = blockIdx.x * blockDim.x + threadIdx.x;
    if (i > N_NODES) return;
    int lo = 0, hi = E_EDGES;
    while (lo < hi) {
        const int mid = (lo + hi) >> 1;
        if (row[mid] < (long long)i) lo = mid + 1; else hi = mid;
    }
    rptr[i] = lo;
}

// ---------------------------------------------------------------------------
// Kernel 3: out[i] = (1/deg_i) * sum_e h[col[e]].  One wave32 per row:
// lane owns float4 of columns (32 lanes x 4 = 128). Each edge is one fully
// coalesced 512B gather from h, served from L2.  Unroll-by-2 with two
// accumulators to keep two gathers in flight per wave.  The streamed-once
// data (col indices, out) use non-temporal hints so they do not evict h
// from L2; the h gathers keep RT hints.
// ---------------------------------------------------------------------------
__global__ __launch_bounds__(256) void aggregate_kernel(
    const long long* __restrict__ col, const int* __restrict__ rptr,
    const float* __restrict__ h, float* __restrict__ out)
{
    const int lane = threadIdx.x & 31;
    const int wave = threadIdx.x >> 5;
    const int rowi = blockIdx.x * 8 + wave;
    if (rowi >= N_NODES) return;

    const int beg = rptr[rowi];
    const int end = rptr[rowi + 1];

    v4f acc0 = {0.f, 0.f, 0.f, 0.f};
    v4f acc1 = {0.f, 0.f, 0.f, 0.f};
    const float* hb = h + 4 * lane;

    int e = beg;
    for (; e + 1 < end; e += 2) {
        __builtin_prefetch(&col[e + 128], 0, 0);          // global_prefetch_b8
        const long long c0 = __builtin_nontemporal_load(&col[e]);
        const long long c1 = __builtin_nontemporal_load(&col[e + 1]);
        const v4f v0 = *(const v4f*)(hb + (size_t)c0 * DIM);
        const v4f v1 = *(const v4f*)(hb + (size_t)c1 * DIM);
        acc0 += v0;
        acc1 += v1;
    }
    if (e < end) {
        const long long c0 = __builtin_nontemporal_load(&col[e]);
        acc0 += *(const v4f*)(hb + (size_t)c0 * DIM);
    }
    acc0 += acc1;

    const float scale = (end > beg) ? 1.0f / (float)(end - beg) : 0.0f;
    v4f r;
    r.x = acc0.x * scale; r.y = acc0.y * scale;
    r.z = acc0.z * scale; r.w = acc0.w * scale;
    __builtin_nontemporal_store(r, (v4f*)(out + (size_t)rowi * DIM + 4 * lane));
}

// ---------------------------------------------------------------------------
// Launch
// ---------------------------------------------------------------------------
extern "C" void kernel_launch(void* const* d_in, const int* in_sizes, int n_in,
                              void* d_out, int out_size, void* d_ws, size_t ws_size,
                              hipStream_t stream)
{
    const float*     x    = (const float*)d_in[0];
    const long long* row  = (const long long*)d_in[1];
    const long long* col  = (const long long*)d_in[2];
    const float*     W    = (const float*)d_in[3];
    const float*     bias = (const float*)d_in[4];
    float*           out  = (float*)d_out;

    char*  ws   = (char*)d_ws;
    float* h    = (float*)ws;                                         // 51.2 MB
    int*   rptr = (int*)(ws + (size_t)N_NODES * DIM * sizeof(float)); // 400 KB

    // h = x @ W + b   (fp32 WMMA; 6250 blocks x 8 waves)
    gemm_bias_wmma<<<N_NODES / 16, 256, 0, stream>>>(x, W, bias, h);

    // CSR pointers from sorted row[]
    rowptr_kernel<<<(N_NODES + 1 + 255) / 256, 256, 0, stream>>>(row, rptr);

    // mean aggregation, one wave per row
    aggregate_kernel<<<(N_NODES + 7) / 8, 256, 0, stream>>>(col, rptr, h, out);
}